// ISSDiagonalizer_32057635897901
// MI455X (gfx1250) — compile-verified
//
#include <hip/hip_runtime.h>
#include <hip/hip_bf16.h>
#include <math.h>

// Problem constants (from reference)
#define BB   8
#define FF   1025
#define MM   8
#define TT   512
#define BF   (BB*FF)            // 8200
#define QFLOATS  ((size_t)2*BB*FF*MM*MM)     // 1,049,600 floats (complex Q interleaved)
#define XTFLOATS ((size_t)BB*FF*MM*TT)       // 33,587,200 floats
#define OUTFLOATS (QFLOATS + XTFLOATS)       // 34,636,800
#define SCALE_DEN 4198400.0f                 // F*M*T = 1025*8*512
#define EPS   1e-6f
#define EPS2  1e-6f
#define EPS3  1e-3f

typedef float v2f __attribute__((ext_vector_type(2)));
typedef float v8f __attribute__((ext_vector_type(8)));

// One block per (b,f). 256 threads = 8 waves (wave32).
__global__ void __launch_bounds__(256)
iss_main(const float* __restrict__ r,
         const float* __restrict__ Qr, const float* __restrict__ Qi,
         const float* __restrict__ xr, const float* __restrict__ xi,
         float* __restrict__ out, float* __restrict__ ws)
{
    const int bf  = blockIdx.x;          // b*F + f
    const int tid = threadIdx.x;
    const size_t xoff = (size_t)bf * (MM*TT);

    const float* rb  = r  + xoff;
    const float* xrb = xr + xoff;
    const float* xib = xi + xoff;

    __shared__ float sP[8*256];          // 8 k-slices of 16x16 P
    __shared__ float sVre[512], sVim[512]; // V[k][m][n] complex
    __shared__ float sQre[64], sQim[64];
    __shared__ float sqre[8],  sqim[8];
    __shared__ float sVqre[64], sVqim[64];
    __shared__ float sqVq[8], svre[8], svim[8];
    __shared__ float sred[256];

    for (int i = tid; i < 8*256; i += 256) sP[i] = 0.0f;
    __syncthreads();

    // ---------------- Stage 1: P_k = (S diag(r_k)) S^T via v_wmma_f32_16x16x4_f32 ----
    {
        const int lane = tid & 31;
        const int wave = tid >> 5;       // 0..7, each wave covers 64 t-columns
        const int mr   = lane & 15;      // row (A) / col (B) index
        const int hi   = lane >> 4;      // lane-half selects K offset {0,2}
        const float* xrow = (mr < 8) ? (xrb + mr*TT) : (xib + (mr-8)*TT);

        v8f acc[8];
#pragma unroll
        for (int k = 0; k < 8; ++k) acc[k] = (v8f){0,0,0,0,0,0,0,0};

        const int tb = wave * 64;
        for (int c = 0; c < 16; ++c) {
            const int t0 = tb + c*4 + hi*2;       // even -> 8B aligned
            v2f u = *(const v2f*)(xrow + t0);     // shared A/B fragment
#pragma unroll
            for (int k = 0; k < 8; ++k) {
                v2f rv = *(const v2f*)(rb + k*TT + t0);
                v2f a;
                a.x = u.x * fmaxf(rv.x, EPS3);
                a.y = u.y * fmaxf(rv.y, EPS3);
                acc[k] = __builtin_amdgcn_wmma_f32_16x16x4_f32(
                    false, a, false, u, (short)0, acc[k], false, false);
            }
        }
        // C/D layout: vgpr j, lanes0-15 -> row j, lanes16-31 -> row j+8; col = lane&15
        const int col = lane & 15;
#pragma unroll
        for (int k = 0; k < 8; ++k)
#pragma unroll
            for (int j = 0; j < 8; ++j)
                atomicAdd(&sP[k*256 + (j + hi*8)*16 + col], acc[k][j]);
    }
    __syncthreads();

    // ---------------- Recombine into complex V, add trace regularizer ----------------
    for (int idx = tid; idx < 512; idx += 256) {
        const int k = idx >> 6, m = (idx >> 3) & 7, n = idx & 7;
        const float* Pk = sP + k*256;
        sVre[idx] = (Pk[m*16 + n] + Pk[(m+8)*16 + (n+8)]) * (1.0f/TT);
        sVim[idx] = (Pk[(m+8)*16 + n] - Pk[m*16 + (n+8)]) * (1.0f/TT);
    }
    if (tid < 64) {
        sQre[tid] = Qr[(size_t)bf*64 + tid];
        sQim[tid] = Qi[(size_t)bf*64 + tid];
    }
    __syncthreads();
    if (tid < 8) {
        float tr = 0.0f;
#pragma unroll
        for (int m = 0; m < 8; ++m) tr += sVre[tid*64 + m*9];
        const float reg = fmaxf(tr, 1.0f) * EPS;
#pragma unroll
        for (int m = 0; m < 8; ++m) sVre[tid*64 + m*9] += reg;
    }
    __syncthreads();

    // ---------------- Stage 2: iterative rank-1 sweep on Q ---------------------------
    for (int it = 0; it < 2; ++it) {
        for (int k = 0; k < 8; ++k) {
            if (tid < 8) { sqre[tid] = sQre[k*8 + tid]; sqim[tid] = sQim[k*8 + tid]; }
            __syncthreads();
            if (tid < 64) {                       // Vq[kk][m] = sum_n V[kk][m][n]*conj(q[n])
                const int kk = tid >> 3, m = tid & 7;
                float re = 0.0f, im = 0.0f;
#pragma unroll
                for (int n = 0; n < 8; ++n) {
                    const float vr = sVre[kk*64 + m*8 + n];
                    const float vi = sVim[kk*64 + m*8 + n];
                    re += vr*sqre[n] + vi*sqim[n];
                    im += vi*sqre[n] - vr*sqim[n];
                }
                sVqre[tid] = re; sVqim[tid] = im;
            }
            __syncthreads();
            if (tid < 8) {                        // qVq, v per kk
                const int kk = tid;
                float qvq = 0.0f, nre = 0.0f, nim = 0.0f;
#pragma unroll
                for (int m = 0; m < 8; ++m) {
                    const float vr = sVqre[kk*8 + m], vi = sVqim[kk*8 + m];
                    qvq += sqre[m]*vr - sqim[m]*vi;
                    nre += sQre[kk*8 + m]*vr - sQim[kk*8 + m]*vi;
                    nim += sQre[kk*8 + m]*vi + sQim[kk*8 + m]*vr;
                }
                qvq = fmaxf(qvq, EPS2);
                sqVq[kk] = qvq;
                if (kk == k) { svre[kk] = 1.0f - 1.0f/sqrtf(qvq); svim[kk] = 0.0f; }
                else         { svre[kk] = nre/qvq;                svim[kk] = nim/qvq; }
            }
            __syncthreads();
            if (tid < 64) {                       // Q[kk][m] -= v[kk]*q[m]
                const int kk = tid >> 3, m = tid & 7;
                sQre[tid] -= svre[kk]*sqre[m] - svim[kk]*sqim[m];
                sQim[tid] -= svre[kk]*sqim[m] + svim[kk]*sqre[m];
            }
            __syncthreads();
        }
    }

    // ---------------- Stage 3: write unscaled Q; compute Qx, xt, per-block sum -------
    if (tid < 64) {
        const size_t qo = ((size_t)bf*64 + tid)*2;
        out[qo]   = sQre[tid];
        out[qo+1] = sQim[tid];
    }

    float lsum = 0.0f;
    for (int t = tid; t < TT; t += 256) {
        float xre[8], xim[8];
#pragma unroll
        for (int n = 0; n < 8; ++n) { xre[n] = xrb[n*TT + t]; xim[n] = xib[n*TT + t]; }
#pragma unroll
        for (int m = 0; m < 8; ++m) {
            float qxr = 0.0f, qxi = 0.0f;
#pragma unroll
            for (int n = 0; n < 8; ++n) {
                const float qr_ = sQre[m*8 + n], qi_ = sQim[m*8 + n];
                qxr += qr_*xre[n] - qi_*xim[n];
                qxi += qr_*xim[n] + qi_*xre[n];
            }
            const float p = qxr*qxr + qxi*qxi;
            out[QFLOATS + (size_t)bf*(MM*TT) + (size_t)m*TT + t] = p;
            lsum += p;
        }
    }
    sred[tid] = lsum;
    __syncthreads();
    for (int off = 128; off > 0; off >>= 1) {
        if (tid < off) sred[tid] += sred[tid + off];
        __syncthreads();
    }
    if (tid == 0) ws[bf] = sred[0];   // deterministic per-block partial sum
}

// One block per b: deterministic reduction of 1025 partial sums -> scale[b]
__global__ void __launch_bounds__(256)
iss_reduce(float* __restrict__ ws)
{
    const int b = blockIdx.x, tid = threadIdx.x;
    __shared__ float s[256];
    float sum = 0.0f;
    for (int f = tid; f < FF; f += 256) sum += ws[b*FF + f];
    s[tid] = sum;
    __syncthreads();
    for (int off = 128; off > 0; off >>= 1) {
        if (tid < off) s[tid] += s[tid + off];
        __syncthreads();
    }
    if (tid == 0) ws[BF + b] = s[0] / SCALE_DEN;
}

// Elementwise normalize: Q /= sqrt(clip(scale,1e-6)); xt /= scale
__global__ void __launch_bounds__(256)
iss_norm(float* __restrict__ out, const float* __restrict__ ws)
{
    const size_t i = (size_t)blockIdx.x*256 + threadIdx.x;
    if (i >= OUTFLOATS) return;
    if (i < QFLOATS) {
        const int b = (int)(i / ((size_t)2*FF*MM*MM));   // / 131200
        const float sc = fmaxf(ws[BF + b], 1e-6f);
        out[i] = out[i] / sqrtf(sc);
    } else {
        const size_t j = i - QFLOATS;
        const int b = (int)(j / ((size_t)FF*MM*TT));     // / 4198400
        out[i] = out[i] / ws[BF + b];
    }
}

extern "C" void kernel_launch(void* const* d_in, const int* in_sizes, int n_in,
                              void* d_out, int out_size, void* d_ws, size_t ws_size,
                              hipStream_t stream) {
    const float* r  = (const float*)d_in[0];
    const float* Qr = (const float*)d_in[1];
    const float* Qi = (const float*)d_in[2];
    const float* xr = (const float*)d_in[3];
    const float* xi = (const float*)d_in[4];
    float* out = (float*)d_out;
    float* ws  = (float*)d_ws;   // [0,BF): block sums; [BF,BF+8): per-b scale

    iss_main<<<BF, 256, 0, stream>>>(r, Qr, Qi, xr, xi, out, ws);
    iss_reduce<<<BB, 256, 0, stream>>>(ws);
    iss_norm<<<(unsigned)((OUTFLOATS + 255)/256), 256, 0, stream>>>(out, ws);
}